// ForecastNetwork_88691074662916
// MI455X (gfx1250) — compile-verified
//
#include <hip/hip_runtime.h>
#include <hip/hip_bf16.h>
#include <math.h>

typedef __attribute__((ext_vector_type(2))) float v2f;
typedef __attribute__((ext_vector_type(8))) float v8f;

#define HID      128
#define ROWS     16     // batch rows per block (one WMMA M-tile)
#define THREADS  128    // 4 waves of 32
#define H1PAD    132    // row stride pad: kills LDS bank conflicts on column reads

__device__ __forceinline__ v8f wmma4(v2f a, v2f b, v8f c) {
  // D = A(16x4 f32) * B(4x16 f32) + C(16x16 f32)
  return __builtin_amdgcn_wmma_f32_16x16x4_f32(false, a, false, b, (short)0, c,
                                               false, false);
}

__device__ __forceinline__ v8f splat8(float x) {
  v8f r;
  #pragma unroll
  for (int i = 0; i < 8; ++i) r[i] = x;
  return r;
}

// Branch-free tanh. Prefer the CDNA5 hardware V_TANH_F32 (single TRANS32 op,
// co-executes with VALU); otherwise a divergence-free exp-based identity
// (plain v_exp_f32; correct at both extremes: exp underflow -> +/-1).
__device__ __forceinline__ float fast_tanh(float x) {
#if __has_builtin(__builtin_amdgcn_tanhf)
  return __builtin_amdgcn_tanhf(x);
#elif __has_builtin(__builtin_amdgcn_tanh_f32)
  return __builtin_amdgcn_tanh_f32(x);
#else
  float t = __expf(-2.0f * fabsf(x));        // in (0,1], branchless
  float r = (1.0f - t) / (1.0f + t);
  return copysignf(r, x);
#endif
}

__global__ __launch_bounds__(THREADS) void forecast_rk45_kernel(
    const float* __restrict__ coords,   // (B,3,3)
    const float* __restrict__ tstamps,  // (B,3)
    const float* __restrict__ W1,       // (6,128)
    const float* __restrict__ b1,       // (128,)
    const float* __restrict__ W2,       // (128,128)
    const float* __restrict__ b2,       // (128,)
    const float* __restrict__ W3,       // (128,6)
    const float* __restrict__ b3,       // (6,)
    const int*   __restrict__ flp,      // forecast_length (scalar)
    float*       __restrict__ out,      // (B,2,3)
    int Bn)
{
  __shared__ float sW1[8][H1PAD];       // K padded 6->8, col-read conflict-free
  __shared__ float sW3[HID][6];
  __shared__ float sb1[HID];
  __shared__ float sb2[HID];
  __shared__ float sb3[6];
  __shared__ float sh1[ROWS][H1PAD];    // layer-1 activations (A-source for layer 2)
  __shared__ float sh2[ROWS][H1PAD];    // layer-2 activations (source for layer 3)
  __shared__ float sxe[ROWS][10];       // eval input state, K padded 6->8(+2)
  __shared__ float sx [ROWS][6];        // integrated state
  __shared__ float skk[6][ROWS][6];     // k1..k6
  __shared__ int   sidx[ROWS][2];
  __shared__ int   smax;

  const int tid  = threadIdx.x;
  const int lane = tid & 31;
  const int w    = tid >> 5;      // wave id 0..3
  const int lm   = lane & 15;     // lane mod 16
  const int half = lane >> 4;     // 0/1 half of wave
  const int b0   = blockIdx.x * ROWS;

  // ---- stage weights into LDS (zero-pad W1's K dim to 8) ----
  for (int i = tid; i < 8 * HID; i += THREADS) {
    int r = i >> 7, c = i & 127;
    sW1[r][c] = (r < 6) ? W1[r * HID + c] : 0.0f;
  }
  for (int i = tid; i < HID; i += THREADS) { sb1[i] = b1[i]; sb2[i] = b2[i]; }
  for (int i = tid; i < HID * 6; i += THREADS) sW3[i / 6][i % 6] = W3[i];
  if (tid < 6) sb3[tid] = b3[tid];
  if (tid < ROWS) { sxe[tid][6] = 0.0f; sxe[tid][7] = 0.0f; sxe[tid][8] = 0.0f; sxe[tid][9] = 0.0f; }

  const float vdt = (float)(1.0 / 15.0 / 2.0);

  // ---- initial state x0 = [coords[:,1], (coords[:,2]-coords[:,0])/(t2'-t0')], sample indices ----
  if (tid == 0) smax = 0;
  if (tid < ROWS) {
    int b = b0 + tid;
    if (b < Bn) {
      float t0 = tstamps[b * 3 + 0], t1 = tstamps[b * 3 + 1], t2 = tstamps[b * 3 + 2];
      float a0 = t0 - t1, a2 = t2 - t1;      // shift-then-diff, as in reference
      float den = a2 - a0;
      #pragma unroll
      for (int c = 0; c < 3; ++c) {
        sx[tid][c]     = coords[b * 9 + 3 + c];
        sx[tid][3 + c] = (coords[b * 9 + 6 + c] - coords[b * 9 + 0 + c]) / den;
      }
      // t_span length from forecast_length (matches jnp.arange count)
      int    fl   = flp[0];
      double stop = (double)(fl + 1) * (1.0 / 15.0) + 0.0001;
      int    n_t  = (int)ceil(stop / (1.0 / 15.0 / 2.0));
      float ts[2] = {0.0f, a2};
      #pragma unroll
      for (int j = 0; j < 2; ++j) {
        float best = __builtin_inff(); int bi = 0;
        for (int i = 0; i < n_t; ++i) {
          float d = fabsf((float)i * vdt - ts[j]);
          if (d < best) { best = d; bi = i; }   // first-min tie-break = argmin
        }
        sidx[tid][j] = bi;
      }
    } else {
      #pragma unroll
      for (int c = 0; c < 6; ++c) sx[tid][c] = 0.0f;
      sidx[tid][0] = -1; sidx[tid][1] = -1;
    }
  }
  __syncthreads();
  if (tid < ROWS) atomicMax(&smax, max(sidx[tid][0], sidx[tid][1]));
  // outputs that sample t_span[0] (state = x0)
  if (tid < 96) {
    int r = tid / 6, c = tid % 6;
    if (c < 3) {
      if (sidx[r][0] == 0) out[(b0 + r) * 6 + c]     = sx[r][c];
      if (sidx[r][1] == 0) out[(b0 + r) * 6 + 3 + c] = sx[r][c];
    }
  }
  __syncthreads();
  const int nsteps = smax;   // integrate only as far as the output needs

  // ---- drift: k[ki] = MLP(sxe) ----
  auto drift = [&](int ki) {
    __syncthreads();                                   // sxe ready
    // ----- layer 1: (16x8) @ (8x128), K in 2 WMMA steps -----
    v2f A0 = *(const v2f*)&sxe[lm][2 * half];          // K = {0,1}+2*half
    v2f A1 = *(const v2f*)&sxe[lm][4 + 2 * half];      // K = {4,5}+2*half
    #pragma unroll
    for (int s = 0; s < 2; ++s) {
      int col = (w * 2 + s) * 16 + lm;
      v8f acc = splat8(sb1[col]);
      v2f B0 = { sW1[2 * half][col],     sW1[2 * half + 1][col] };
      acc = wmma4(A0, B0, acc);
      v2f B1 = { sW1[4 + 2 * half][col], sW1[4 + 2 * half + 1][col] };
      acc = wmma4(A1, B1, acc);
      #pragma unroll
      for (int v = 0; v < 8; ++v) sh1[v + 8 * half][col] = fast_tanh(acc[v]);
    }
    __syncthreads();                                   // sh1 ready
    // ----- layer 2: (16x128) @ (128x128), 32 K-steps, 2 N-tiles/wave -----
    {
      int col0 = (w * 2 + 0) * 16 + lm;
      int col1 = (w * 2 + 1) * 16 + lm;
      v8f acc0 = splat8(sb2[col0]);
      v8f acc1 = splat8(sb2[col1]);
      #pragma unroll 4
      for (int k0 = 0; k0 < 32; ++k0) {
        int kA = k0 * 4 + 2 * half;
        v2f A  = *(const v2f*)&sh1[lm][kA];            // A-frag reused for both N-tiles
        v2f Bf0 = { W2[kA * HID + col0], W2[(kA + 1) * HID + col0] };
        acc0 = wmma4(A, Bf0, acc0);
        v2f Bf1 = { W2[kA * HID + col1], W2[(kA + 1) * HID + col1] };
        acc1 = wmma4(A, Bf1, acc1);
      }
      #pragma unroll
      for (int v = 0; v < 8; ++v) {
        sh2[v + 8 * half][col0] = fast_tanh(acc0[v]);
        sh2[v + 8 * half][col1] = fast_tanh(acc1[v]);
      }
    }
    __syncthreads();                                   // sh2 ready
    // ----- layer 3: (16x128) @ (128x6) via VALU (4.5% of FLOPs) -----
    if (tid < 96) {
      int r = tid / 6, c = tid % 6;
      float acc = 0.0f;
      #pragma unroll 8
      for (int k = 0; k < HID; ++k) acc = fmaf(sh2[r][k], sW3[k][c], acc);
      skk[ki][r][c] = acc + sb3[c];
    }
    __syncthreads();                                   // skk ready
  };

  auto stage = [&](float dt, int n, const float* cs) {
    if (tid < 96) {
      int r = tid / 6, c = tid % 6;
      float s = 0.0f;
      for (int i = 0; i < n; ++i) s += cs[i] * skk[i][r][c];
      sxe[r][c] = sx[r][c] + dt * s;
    }
  };

  for (int ti = 0; ti < nsteps; ++ti) {
    float dt = (float)(ti + 1) * vdt - (float)ti * vdt;   // diff of fp32 t_span
    if (tid < 96) { int r = tid / 6, c = tid % 6; sxe[r][c] = sx[r][c]; }
    drift(0);
    { const float cs[1] = {(float)(1.0 / 5.0)}; stage(dt, 1, cs); }
    drift(1);
    { const float cs[2] = {(float)(3.0 / 40.0), (float)(9.0 / 40.0)}; stage(dt, 2, cs); }
    drift(2);
    { const float cs[3] = {(float)(44.0 / 45.0), (float)(-56.0 / 15.0), (float)(32.0 / 9.0)};
      stage(dt, 3, cs); }
    drift(3);
    { const float cs[4] = {(float)(19372.0 / 6561.0), (float)(-25360.0 / 2187.0),
                           (float)(64448.0 / 6561.0), (float)(-212.0 / 729.0)};
      stage(dt, 4, cs); }
    drift(4);
    { const float cs[5] = {(float)(9017.0 / 3168.0), (float)(-355.0 / 33.0),
                           (float)(46732.0 / 5247.0), (float)(49.0 / 176.0),
                           (float)(-5103.0 / 18656.0)};
      stage(dt, 5, cs); }
    drift(5);
    // xn = x + dt*(b1*k1 + b3*k3 + b4*k4 + b5*k5 + b6*k6)
    if (tid < 96) {
      int r = tid / 6, c = tid % 6;
      float s = (float)(35.0 / 384.0) * skk[0][r][c];
      s += (float)(500.0 / 1113.0)  * skk[2][r][c];
      s += (float)(125.0 / 192.0)   * skk[3][r][c];
      s += (float)(-2187.0 / 6784.0) * skk[4][r][c];
      s += (float)(11.0 / 84.0)     * skk[5][r][c];
      float xn = sx[r][c] + dt * s;
      sx[r][c] = xn;
      if (c < 3) {
        if (sidx[r][0] == ti + 1) out[(b0 + r) * 6 + c]     = xn;
        if (sidx[r][1] == ti + 1) out[(b0 + r) * 6 + 3 + c] = xn;
      }
    }
    // next iteration's drift(0) entry barrier orders all cross-thread deps
  }
}

extern "C" void kernel_launch(void* const* d_in, const int* in_sizes, int n_in,
                              void* d_out, int out_size, void* d_ws, size_t ws_size,
                              hipStream_t stream) {
  const float* coords  = (const float*)d_in[0];
  const float* tstamps = (const float*)d_in[1];
  const float* W1 = (const float*)d_in[2];
  const float* b1 = (const float*)d_in[3];
  const float* W2 = (const float*)d_in[4];
  const float* b2 = (const float*)d_in[5];
  const float* W3 = (const float*)d_in[6];
  const float* b3 = (const float*)d_in[7];
  const int*   fl = (const int*)d_in[8];
  float* out = (float*)d_out;
  int Bn = in_sizes[0] / 9;
  int blocks = (Bn + ROWS - 1) / ROWS;
  forecast_rk45_kernel<<<blocks, THREADS, 0, stream>>>(
      coords, tstamps, W1, b1, W2, b2, W3, b3, fl, out, Bn);
}